// LearnedPooling_42803644072775
// MI455X (gfx1250) — compile-verified
//
#include <hip/hip_runtime.h>
#include <math.h>

typedef __attribute__((ext_vector_type(2))) float v2f;
typedef __attribute__((ext_vector_type(8))) float v8f;

#define N_PTS   262144
#define C_DIM   256
#define P_POOL  3
#define B_SEG   64

#define K1_BLOCKS       2048   // 128 rows per block (8 waves x 16 rows)
#define ROWS_PER_BLOCK4 128
#define K4_BLOCKS       (N_PTS / ROWS_PER_BLOCK4)

// workspace layout (float offsets); total ~3.2 MB
#define WS_SCORES 0
#define WS_PM     (N_PTS * 3)
#define WS_PS     (WS_PM + K1_BLOCKS * 3)
#define WS_FIN    (WS_PS + K1_BLOCKS * 3)   // 6 floats: m[3], invZ[3]

#define NEG_BIG (-3.402823466e+38f)

// ---------------------------------------------------------------------------
// Kernel 1: scores = x @ W^T + b via V_WMMA_F32_16X16X4_F32, plus per-block
// online-softmax partials (m, s) per pool.
// Block: 256 threads (8 waves), 128 rows. Grid: 2048.
// ---------------------------------------------------------------------------
__global__ __launch_bounds__(256)
void scores_wmma_kernel(const float* __restrict__ x,
                        const float* __restrict__ pw,   // [P, C] row-major
                        const float* __restrict__ pb,   // [P]
                        float* __restrict__ scores,     // [N, 3]
                        float* __restrict__ partM,      // [K1_BLOCKS, 3]
                        float* __restrict__ partS) {    // [K1_BLOCKS, 3]
    __shared__ float Bl[16 * C_DIM];   // zero-padded W^T: Bl[n*256 + k]
    __shared__ float redM[P_POOL][16];
    __shared__ float redS[P_POOL][16];

    const int tid = threadIdx.x;
    for (int j = tid; j < 16 * C_DIM; j += 256)
        Bl[j] = (j < P_POOL * C_DIM) ? pw[j] : 0.0f;
    __syncthreads();

    const int wave = tid >> 5;
    const int lane = tid & 31;
    const int m    = lane & 15;   // column (pool) / row-in-tile id
    const int hi   = lane >> 4;   // 0: K=0,1  1: K=2,3 (A & B ISA layout)
    const int koff = hi * 2;
    const int r0   = blockIdx.x * 128 + wave * 16;

    const float* xrow = x + (size_t)(r0 + m) * C_DIM;
    const float* brow = &Bl[m * C_DIM];

    v8f acc = {0.f, 0.f, 0.f, 0.f, 0.f, 0.f, 0.f, 0.f};
    for (int k = 0; k < C_DIM; k += 4) {
        v2f a = *(const v2f*)(xrow + k + koff);   // A[M=m][K=koff..koff+1]
        v2f b = *(const v2f*)(brow + k + koff);   // B[K=koff..koff+1][N=m]
        acc = __builtin_amdgcn_wmma_f32_16x16x4_f32(
            /*neg_a=*/false, a, /*neg_b=*/false, b,
            /*c_mod=*/(short)0, acc, /*reuse_a=*/false, /*reuse_b=*/false);
    }

    // D layout: VGPR j -> row r0 + hi*8 + j, col = m (valid for m < 3)
    const float bias = (m < P_POOL) ? pb[m] : 0.0f;
    float mm = NEG_BIG, ss = 0.0f;
#pragma unroll
    for (int j = 0; j < 8; ++j) {
        float s = acc[j] + bias;
        acc[j] = s;
        mm = fmaxf(mm, s);
    }
#pragma unroll
    for (int j = 0; j < 8; ++j) ss += __expf(acc[j] - mm);

    if (m < P_POOL) {
        const int rbase = r0 + hi * 8;
#pragma unroll
        for (int j = 0; j < 8; ++j)
            scores[(size_t)(rbase + j) * 3 + m] = acc[j];
        redM[m][wave * 2 + hi] = mm;
        redS[m][wave * 2 + hi] = ss;
    }
    __syncthreads();

    if (tid < P_POOL) {
        float M = NEG_BIG, S = 0.0f;
#pragma unroll
        for (int i = 0; i < 16; ++i) {
            float mi = redM[tid][i], si = redS[tid][i];
            float M2 = fmaxf(M, mi);
            S = S * __expf(M - M2) + si * __expf(mi - M2);
            M = M2;
        }
        partM[blockIdx.x * 3 + tid] = M;
        partS[blockIdx.x * 3 + tid] = S;
    }
}

// ---------------------------------------------------------------------------
// Kernel 2: merge block partials -> global m_p and 1/Z_p. One block, 96
// threads = 3 wave32's, one wave per pool; wave32 shuffle tree merge.
// ---------------------------------------------------------------------------
__global__ __launch_bounds__(96)
void reduce_partials_kernel(const float* __restrict__ partM,
                            const float* __restrict__ partS,
                            float* __restrict__ fin) {
    const int p    = threadIdx.x >> 5;
    const int lane = threadIdx.x & 31;
    float M = NEG_BIG, S = 0.0f;
    for (int i = lane; i < K1_BLOCKS; i += 32) {
        float mi = partM[i * 3 + p], si = partS[i * 3 + p];
        float M2 = fmaxf(M, mi);
        S = S * __expf(M - M2) + si * __expf(mi - M2);
        M = M2;
    }
#pragma unroll
    for (int off = 16; off > 0; off >>= 1) {
        float mo = __shfl_xor(M, off, 32);
        float so = __shfl_xor(S, off, 32);
        float M2 = fmaxf(M, mo);
        S = S * __expf(M - M2) + so * __expf(mo - M2);
        M = M2;
    }
    if (lane == 0) {
        fin[p]     = M;
        fin[3 + p] = 1.0f / S;
    }
}

// ---------------------------------------------------------------------------
// Kernel 3: scores -> att in place: att[i,p] = exp(s - m_p) * invZ_p
// ---------------------------------------------------------------------------
__global__ __launch_bounds__(256)
void att_kernel(float* __restrict__ sc, const float* __restrict__ fin) {
    const int i = blockIdx.x * 256 + threadIdx.x;
    if (i >= N_PTS) return;
    const float m0 = fin[0], m1 = fin[1], m2 = fin[2];
    const float z0 = fin[3], z1 = fin[4], z2 = fin[5];
    float s0 = sc[(size_t)i * 3 + 0];
    float s1 = sc[(size_t)i * 3 + 1];
    float s2 = sc[(size_t)i * 3 + 2];
    sc[(size_t)i * 3 + 0] = __expf(s0 - m0) * z0;
    sc[(size_t)i * 3 + 1] = __expf(s1 - m1) * z1;
    sc[(size_t)i * 3 + 2] = __expf(s2 - m2) * z2;
}

// ---------------------------------------------------------------------------
// Kernel 4: zero output (atomics accumulate into it)
// ---------------------------------------------------------------------------
__global__ __launch_bounds__(256)
void zero_kernel(float* __restrict__ out, int n) {
    const int i = blockIdx.x * 256 + threadIdx.x;
    if (i < n) out[i] = 0.0f;
}

// ---------------------------------------------------------------------------
// Kernel 5: segment-sum pass. Thread = channel c, block = 128 consecutive
// rows. batch is sorted -> ~1-2 segments per block; register accumulate,
// flush with f32 atomics on segment change.
// out[b*768 + p*256 + c]
// ---------------------------------------------------------------------------
__global__ __launch_bounds__(256)
void pool_kernel(const float* __restrict__ x,
                 const int*   __restrict__ batch,
                 const float* __restrict__ att,
                 float* __restrict__ out) {
    const int c  = threadIdx.x;
    const int r0 = blockIdx.x * ROWS_PER_BLOCK4;
    int curb = batch[r0];
    float a0 = 0.f, a1 = 0.f, a2 = 0.f;
    for (int r = r0; r < r0 + ROWS_PER_BLOCK4; ++r) {
        const int b = batch[r];
        if (b != curb) {
            atomicAdd(&out[curb * 768 + 0 * 256 + c], a0);
            atomicAdd(&out[curb * 768 + 1 * 256 + c], a1);
            atomicAdd(&out[curb * 768 + 2 * 256 + c], a2);
            a0 = a1 = a2 = 0.f;
            curb = b;
        }
        const float xv = x[(size_t)r * C_DIM + c];
        const float w0 = att[(size_t)r * 3 + 0];
        const float w1 = att[(size_t)r * 3 + 1];
        const float w2 = att[(size_t)r * 3 + 2];
        a0 = fmaf(w0, xv, a0);
        a1 = fmaf(w1, xv, a1);
        a2 = fmaf(w2, xv, a2);
    }
    atomicAdd(&out[curb * 768 + 0 * 256 + c], a0);
    atomicAdd(&out[curb * 768 + 1 * 256 + c], a1);
    atomicAdd(&out[curb * 768 + 2 * 256 + c], a2);
}

// ---------------------------------------------------------------------------
extern "C" void kernel_launch(void* const* d_in, const int* in_sizes, int n_in,
                              void* d_out, int out_size, void* d_ws, size_t ws_size,
                              hipStream_t stream) {
    const float* x  = (const float*)d_in[0];
    const int*   bt = (const int*)d_in[1];
    const float* pw = (const float*)d_in[2];
    const float* pb = (const float*)d_in[3];
    float* out = (float*)d_out;
    float* ws  = (float*)d_ws;

    float* scores = ws + WS_SCORES;   // N*3, overwritten in-place with att
    float* partM  = ws + WS_PM;
    float* partS  = ws + WS_PS;
    float* fin    = ws + WS_FIN;

    scores_wmma_kernel<<<K1_BLOCKS, 256, 0, stream>>>(x, pw, pb, scores, partM, partS);
    reduce_partials_kernel<<<1, 96, 0, stream>>>(partM, partS, fin);
    att_kernel<<<N_PTS / 256, 256, 0, stream>>>(scores, fin);
    zero_kernel<<<(out_size + 255) / 256, 256, 0, stream>>>(out, out_size);
    pool_kernel<<<K4_BLOCKS, 256, 0, stream>>>(x, bt, scores, out);
}